// HeadBlock_19628000542746
// MI455X (gfx1250) — compile-verified
//
#include <hip/hip_runtime.h>
#include <hip/hip_bf16.h>

typedef float v2f __attribute__((ext_vector_type(2)));
typedef float v4f __attribute__((ext_vector_type(4)));
typedef float v8f __attribute__((ext_vector_type(8)));

// ---------------------------------------------------------------------------
// Kernel 1: 1x1 conv (GEMM 1024x1024x102 via V_WMMA_F32_16X16X4_F32)
//           + bias + depth-to-space fused into the store epilogue.
// Output: img[b][I][J], b in [0,16), I,J in [0,256).  (4 MB, stays in L2)
// y[b, m, i, j] = sum_k x[b,k,i,j] * Wup[m,k] + bup[m]
// img[b, i*32 + (m>>5), j*32 + (m&31)] = y[b, m, i, j]
// ---------------------------------------------------------------------------
__global__ __launch_bounds__(128) void upconv_d2s_wmma(
    const float* __restrict__ x,    // [16][102][8][8]
    const float* __restrict__ Wup,  // [1024][102]
    const float* __restrict__ bup,  // [1024]
    float* __restrict__ img)        // [16][256][256]
{
    const int wave = threadIdx.x >> 5;
    const int lane = threadIdx.x & 31;
    const int tile = blockIdx.x * 4 + wave;          // 4096 tiles
    const int m0   = (tile >> 6) << 4;               // output-channel tile
    const int n0   = (tile & 63) << 4;               // position tile (b*64+ij)
    const int half = lane >> 4;                      // K-pair select
    const int l15  = lane & 15;

    // A: row m = m0 + l15, K values kb, kb+1 (kb = k + 2*half)
    const float* wrow = Wup + (m0 + l15) * 102;
    // B: column n = n0 + l15 -> (b, ij); K values at stride 64 floats
    const int nB  = n0 + l15;
    const int bB  = nB >> 6;
    const int ijB = nB & 63;
    const float* xcol = x + bB * (102 * 64) + ijB;

    v8f acc = {};
    for (int k = 0; k < 104; k += 4) {               // K padded 102 -> 104
        const int kb = k + 2 * half;                 // kb is even
        v2f a, b;
        if (kb < 102) {                              // kb<102 => kb+1<=101 valid
            a.x = wrow[kb];
            a.y = wrow[kb + 1];
            b.x = xcol[kb * 64];
            b.y = xcol[(kb + 1) * 64];
        } else {
            a = (v2f){0.0f, 0.0f};
            b = (v2f){0.0f, 0.0f};
        }
        acc = __builtin_amdgcn_wmma_f32_16x16x4_f32(
            /*neg_a=*/false, a, /*neg_b=*/false, b,
            /*c_mod=*/(short)0, acc, /*reuse_a=*/false, /*reuse_b=*/false);
    }

    // Epilogue: bias + depth-to-space scatter. C/D layout: vgpr r holds
    // M = r + 8*half, lanes l15 hold N = n0 + l15.
    const int i = ijB >> 3;
    const int j = ijB & 7;
    float* bimg = img + bB * 65536;
#pragma unroll
    for (int r = 0; r < 8; ++r) {
        const int m = m0 + r + 8 * half;
        const float v = acc[r] + bup[m];
        const int I = (i << 5) + (m >> 5);
        const int J = (j << 5) + (m & 31);
        bimg[I * 256 + J] = v;
    }
}

// ---------------------------------------------------------------------------
// Kernel 2: 3x3 SAME conv 1->64 + bias + PReLU, single pass.
// Each thread: 4 consecutive J pixels for one (b, I); 3x6 halo in registers,
// channel loop inner; weights/bias/slope in LDS; NT float4 stores (256 MB
// written exactly once, never re-read).
// ---------------------------------------------------------------------------
__global__ __launch_bounds__(256) void conv3_prelu(
    const float* __restrict__ img,  // [16][256][256]
    const float* __restrict__ W1,   // [64][9]
    const float* __restrict__ b1,   // [64]
    const float* __restrict__ a,    // [1]
    float* __restrict__ z)          // [16][64][256][256]
{
    __shared__ float sW[64 * 9];
    __shared__ float sB[64];
    __shared__ float sA;

    const int tid = threadIdx.x;
    for (int t = tid; t < 576; t += 256) sW[t] = W1[t];
    if (tid < 64) sB[tid] = b1[tid];
    if (tid == 0) sA = a[0];
    __syncthreads();

    const int b  = blockIdx.x >> 6;
    const int I  = ((blockIdx.x & 63) << 2) + (tid >> 6);  // 4 rows per block
    const int Jb = (tid & 63) << 2;                         // 4 pixels per thread
    const float* base = img + b * 65536;

    float nb[3][6];
#pragma unroll
    for (int dI = 0; dI < 3; ++dI) {
        const int row = I + dI - 1;
        const bool rv = (row >= 0) && (row < 256);
        const float* rp = base + row * 256;
        v4f c4;
        if (rv) c4 = *(const v4f*)(rp + Jb);
        else    c4 = (v4f){0.f, 0.f, 0.f, 0.f};
        nb[dI][1] = c4.x; nb[dI][2] = c4.y; nb[dI][3] = c4.z; nb[dI][4] = c4.w;
        nb[dI][0] = (rv && Jb > 0)         ? rp[Jb - 1] : 0.f;
        nb[dI][5] = (rv && (Jb + 4) < 256) ? rp[Jb + 4] : 0.f;
    }

    const float slope = sA;
    float* zb = z + ((size_t)b * 64) * 65536 + I * 256 + Jb;

    for (int c = 0; c < 64; ++c) {
        const float* w = &sW[c * 9];
        const float bias = sB[c];
        float o0 = bias, o1 = bias, o2 = bias, o3 = bias;
#pragma unroll
        for (int dI = 0; dI < 3; ++dI) {
#pragma unroll
            for (int dJ = 0; dJ < 3; ++dJ) {
                const float wt = w[dI * 3 + dJ];
                o0 = fmaf(nb[dI][0 + dJ], wt, o0);
                o1 = fmaf(nb[dI][1 + dJ], wt, o1);
                o2 = fmaf(nb[dI][2 + dJ], wt, o2);
                o3 = fmaf(nb[dI][3 + dJ], wt, o3);
            }
        }
        v4f r;
        r.x = (o0 >= 0.f) ? o0 : slope * o0;
        r.y = (o1 >= 0.f) ? o1 : slope * o1;
        r.z = (o2 >= 0.f) ? o2 : slope * o2;
        r.w = (o3 >= 0.f) ? o3 : slope * o3;
        __builtin_nontemporal_store(r, (v4f*)(zb + (size_t)c * 65536));
    }
}

extern "C" void kernel_launch(void* const* d_in, const int* in_sizes, int n_in,
                              void* d_out, int out_size, void* d_ws, size_t ws_size,
                              hipStream_t stream) {
    const float* x   = (const float*)d_in[0];  // [16,102,8,8]
    const float* Wup = (const float*)d_in[1];  // [1024,102,1,1]
    const float* bup = (const float*)d_in[2];  // [1024]
    const float* W1  = (const float*)d_in[3];  // [64,1,3,3]
    const float* b1  = (const float*)d_in[4];  // [64]
    const float* a   = (const float*)d_in[5];  // [1]
    float* img = (float*)d_ws;                 // 16*256*256 f32 = 4 MB scratch
    float* z   = (float*)d_out;                // [16,64,256,256]

    // 4096 WMMA tiles, 4 waves (128 threads) per block
    upconv_d2s_wmma<<<1024, 128, 0, stream>>>(x, Wup, bup, img);
    // 16 batches * 64 row-groups; 4 rows x 256 cols per block
    conv3_prelu<<<1024, 256, 0, stream>>>(img, W1, b1, a, z);
}